// GAT_10093173145739
// MI455X (gfx1250) — compile-verified
//
#include <hip/hip_runtime.h>
#include <stdint.h>

typedef __attribute__((ext_vector_type(2))) float v2f;
typedef __attribute__((ext_vector_type(8))) float v8f;

#define NEG_SLOPE 0.2f
#define SOFT_EPS 1e-16f

// ---- order-preserving float<->uint encoding for atomicMax-based segment max ----
__device__ __forceinline__ unsigned f2ord(float f) {
  unsigned u = __float_as_uint(f);
  return (u & 0x80000000u) ? ~u : (u | 0x80000000u);
}
__device__ __forceinline__ float ord2f(unsigned u) {
  return (u & 0x80000000u) ? __uint_as_float(u & 0x7fffffffu) : __uint_as_float(~u);
}

// ---- gfx1250 async global->LDS copy (bypasses VGPRs, tracked by ASYNCcnt) ----
__device__ __forceinline__ void async_copy_b32(unsigned lds_off, const float* gptr) {
  asm volatile("global_load_async_to_lds_b32 %0, %1, off"
               :
               : "v"(lds_off), "v"((unsigned long long)(size_t)gptr)
               : "memory");
}
__device__ __forceinline__ void async_wait0() {
  asm volatile("s_wait_asynccnt 0" ::: "memory");
}

// ---------------- fills ----------------
__global__ void fill_f32(float* p, float v, long long n) {
  long long i = blockIdx.x * (long long)blockDim.x + threadIdx.x;
  if (i < n) p[i] = v;
}
__global__ void fill_u32(unsigned* p, unsigned v, long long n) {
  long long i = blockIdx.x * (long long)blockDim.x + threadIdx.x;
  if (i < n) p[i] = v;
}

// ---------------- fp32 WMMA GEMM with LDS-staged B ----------------
// C[M,Nn] = A[M,K] * B[K,Nn].  One wave owns a 16-row stripe across NT 16-col
// tiles (NT*16 == Nn).  B is staged chunk-by-chunk (kchunk rows) into LDS by
// the whole workgroup via async global->LDS copies, pair-packed so rows
// (k, k+1) sit in one float2 => hot loop reads B with single ds_load_b64.
// A 16x4 fp32 layout (ISA 7.12.2): lanes 0-15 hold K={k,k+1}, 16-31 K={k+2,k+3}.
// C/D 16x16 fp32: VGPR r = row M=r (lanes 0-15) / M=r+8 (lanes 16-31), N=lane%16.
template <int NT>
__global__ void gemm_wmma_f32_lds(const float* __restrict__ A, const float* __restrict__ B,
                                  float* __restrict__ C, int M, int K, int Nn, int kchunk) {
  extern __shared__ v2f sB[]; // (kchunk/2) * Nn entries
  const int tid = threadIdx.x;
  const int lane = tid & 31;
  const int half = lane >> 4; // 0: K pair {k,k+1}, 1: {k+2,k+3}
  const int lrow = lane & 15;
  const int mtiles = M >> 4;
  const long long gwave = blockIdx.x * (long long)(blockDim.x >> 5) + (tid >> 5);
  const bool active = gwave < (long long)mtiles; // wave-uniform
  const int row0 = (int)(gwave << 4);
  const unsigned ldsbase = (unsigned)(size_t)sB; // low 32 bits = LDS byte offset

  v8f acc[NT] = {};
  const float* arow = A + (size_t)((active ? row0 : 0) + lrow) * K + 2 * half;

  for (int kb = 0; kb < K; kb += kchunk) {
    __syncthreads(); // protect LDS from previous chunk's readers
    // ---- whole-block async preload of B rows [kb, kb+kchunk), pair-packed ----
    const int entries = (kchunk >> 1) * Nn;
    for (int idx = tid; idx < entries; idx += blockDim.x) {
      int pr = idx / Nn;
      int col = idx - pr * Nn;
      const float* g0 = B + (size_t)(kb + 2 * pr) * Nn + col;
      unsigned loff = ldsbase + (unsigned)idx * 8u;
      async_copy_b32(loff, g0);          // even row -> .x
      async_copy_b32(loff + 4u, g0 + Nn); // odd row  -> .y
    }
    async_wait0();
    __syncthreads();

    if (active) {
      if (kb + kchunk < K) __builtin_prefetch(arow + kb + kchunk, 0, 0);
      for (int k = 0; k < kchunk; k += 4) {
        v2f a;
        a.x = arow[kb + k];
        a.y = arow[kb + k + 1];
        const v2f* bp = sB + ((k + 2 * half) >> 1) * Nn + lrow;
#pragma unroll
        for (int t = 0; t < NT; ++t) {
          v2f b = bp[t * 16]; // ds_load_b64
          acc[t] = __builtin_amdgcn_wmma_f32_16x16x4_f32(
              /*neg_a=*/false, a, /*neg_b=*/false, b,
              /*c_mod=*/(short)0, acc[t], /*reuse_a=*/false, /*reuse_b=*/false);
        }
      }
    }
  }

  if (!active) return;
#pragma unroll
  for (int t = 0; t < NT; ++t) {
#pragma unroll
    for (int r = 0; r < 8; ++r) {
      C[(size_t)(row0 + r + 8 * half) * Nn + t * 16 + lrow] = acc[t][r];
    }
  }
}

// ---------------- per-node attention coefficients ----------------
__global__ void att_coeff(const float* __restrict__ h, const float* __restrict__ att_src,
                          const float* __restrict__ att_dst, float* __restrict__ asrc,
                          float* __restrict__ adst, int n_nodes, int H, int C) {
  long long idx = blockIdx.x * (long long)blockDim.x + threadIdx.x;
  if (idx >= (long long)n_nodes * H) return;
  int hd = (int)(idx % H);
  long long n = idx / H;
  const float* hp = h + (size_t)n * H * C + (size_t)hd * C;
  const float* as = att_src + (size_t)hd * C;
  const float* ad = att_dst + (size_t)hd * C;
  float s = 0.f, d = 0.f;
  for (int c = 0; c < C; ++c) {
    float v = hp[c];
    s += v * as[c];
    d += v * ad[c];
  }
  asrc[idx] = s;
  adst[idx] = d;
}

// ---------------- edge pass 1: segment max of leaky-relu(e) over dst ----------------
__global__ void edge_max(const long long* __restrict__ ei, long long E, long long ET, int H,
                         const float* __restrict__ asrc, const float* __restrict__ adst,
                         unsigned* __restrict__ m) {
  long long idx = blockIdx.x * (long long)blockDim.x + threadIdx.x;
  if (idx >= ET * H) return;
  long long e = idx / H;
  int hd = (int)(idx % H);
  long long s = (e < E) ? ei[e] : (e - E); // self loops appended
  long long d = (e < E) ? ei[E + e] : (e - E);
  float v = asrc[s * H + hd] + adst[d * H + hd];
  v = (v >= 0.f) ? v : NEG_SLOPE * v;
  atomicMax(&m[d * H + hd], f2ord(v));
}

// ---------------- edge pass 2: segment sum of exp(e - max) ----------------
__global__ void edge_sum(const long long* __restrict__ ei, long long E, long long ET, int H,
                         const float* __restrict__ asrc, const float* __restrict__ adst,
                         const unsigned* __restrict__ m, float* __restrict__ den) {
  long long idx = blockIdx.x * (long long)blockDim.x + threadIdx.x;
  if (idx >= ET * H) return;
  long long e = idx / H;
  int hd = (int)(idx % H);
  long long s = (e < E) ? ei[e] : (e - E);
  long long d = (e < E) ? ei[E + e] : (e - E);
  float v = asrc[s * H + hd] + adst[d * H + hd];
  v = (v >= 0.f) ? v : NEG_SLOPE * v;
  float ex = __expf(v - ord2f(m[d * H + hd]));
  atomicAdd(&den[d * H + hd], ex);
}

// ---------------- edge pass 3: alpha-weighted gather/scatter of features ----------------
// F = H*C consecutive threads own one edge: coalesced gathers + float atomics.
template <int F, int C>
__global__ void edge_aggregate(const long long* __restrict__ ei, long long E, long long ET,
                               int H, const float* __restrict__ asrc,
                               const float* __restrict__ adst, const unsigned* __restrict__ m,
                               const float* __restrict__ den, const float* __restrict__ hfeat,
                               float* __restrict__ out) {
  const int epb = 256 / F;
  long long e = blockIdx.x * (long long)epb + threadIdx.x / F;
  if (e >= ET) return;
  int f = threadIdx.x % F;
  int hd = f / C;
  long long s = (e < E) ? ei[e] : (e - E);
  long long d = (e < E) ? ei[E + e] : (e - E);
  float v = asrc[s * H + hd] + adst[d * H + hd];
  v = (v >= 0.f) ? v : NEG_SLOPE * v;
  float ex = __expf(v - ord2f(m[d * H + hd]));
  float alpha = ex / (den[d * H + hd] + SOFT_EPS);
  atomicAdd(&out[d * (size_t)F + f], hfeat[s * (size_t)F + f] * alpha);
}

// ---------------- elementwise ----------------
__global__ void elu_bias(float* __restrict__ x, const float* __restrict__ b, long long n, int F) {
  long long i = blockIdx.x * (long long)blockDim.x + threadIdx.x;
  if (i >= n) return;
  float v = x[i] + b[i % F];
  x[i] = (v > 0.f) ? v : (__expf(v) - 1.f);
}

__global__ void add_bias(float* __restrict__ x, const float* __restrict__ b, long long n, int F) {
  long long i = blockIdx.x * (long long)blockDim.x + threadIdx.x;
  if (i >= n) return;
  x[i] = x[i] + b[i % F];
}

static inline long long cdiv(long long a, long long b) { return (a + b - 1) / b; }

extern "C" void kernel_launch(void* const* d_in, const int* in_sizes, int n_in,
                              void* d_out, int out_size, void* d_ws, size_t ws_size,
                              hipStream_t stream) {
  // inputs (setup_inputs order)
  const float* x = (const float*)d_in[0];          // [N,256]
  const long long* ei = (const long long*)d_in[1]; // [2,E] int64
  const float* W1 = (const float*)d_in[2];         // [256,128]
  const float* att_src1 = (const float*)d_in[3];   // [4,32]
  const float* att_dst1 = (const float*)d_in[4];   // [4,32]
  const float* b1 = (const float*)d_in[5];         // [128]
  const float* W2 = (const float*)d_in[6];         // [128,64]
  const float* att_src2 = (const float*)d_in[7];   // [1,64]
  const float* att_dst2 = (const float*)d_in[8];   // [1,64]
  const float* b2 = (const float*)d_in[9];         // [64]
  float* out = (float*)d_out;                      // [N,64]

  const int Fin = 256, H1 = 4, C1 = 32, F1 = 128, H2 = 1, C2 = 64, F2 = 64;
  const long long N = in_sizes[0] / Fin;
  const long long E = in_sizes[1] / 2;
  const long long ET = E + N; // with self loops

  // workspace partition (floats)
  float* ws = (float*)d_ws;
  float* h1 = ws;                    // N*128
  float* out1 = h1 + N * F1;         // N*128 (becomes ELU activation in place)
  float* h2 = out1 + N * F1;         // N*64
  float* asrc1 = h2 + N * F2;        // N*4
  float* adst1 = asrc1 + N * H1;     // N*4
  unsigned* m1 = (unsigned*)(adst1 + N * H1); // N*4
  float* den1 = (float*)m1 + N * H1; // N*4
  float* asrc2 = den1 + N * H1;      // N
  float* adst2 = asrc2 + N * H2;     // N
  unsigned* m2 = (unsigned*)(adst2 + N * H2); // N
  float* den2 = (float*)m2 + N * H2; // N

  const int TB = 256;
  const int KCHUNK = 128; // LDS staging: (128/2)*Nn float2 => 64KB (Nn=128) / 32KB (Nn=64)

  // ---- init accumulators ----
  fill_u32<<<cdiv(N * H1, TB), TB, 0, stream>>>(m1, 0u, N * H1);
  fill_f32<<<cdiv(N * H1, TB), TB, 0, stream>>>(den1, 0.f, N * H1);
  fill_f32<<<cdiv(N * F1, TB), TB, 0, stream>>>(out1, 0.f, N * F1);
  fill_u32<<<cdiv(N * H2, TB), TB, 0, stream>>>(m2, 0u, N * H2);
  fill_f32<<<cdiv(N * H2, TB), TB, 0, stream>>>(den2, 0.f, N * H2);
  fill_f32<<<cdiv(N * F2, TB), TB, 0, stream>>>(out, 0.f, N * F2);

  // ---- layer 1 ----
  {
    long long waves = N / 16; // N = 100000 = 16*6250
    size_t shmem = (size_t)(KCHUNK / 2) * F1 * sizeof(v2f); // 64KB
    gemm_wmma_f32_lds<8><<<cdiv(waves, TB / 32), TB, shmem, stream>>>(
        x, W1, h1, (int)N, Fin, F1, KCHUNK);
  }
  att_coeff<<<cdiv(N * H1, TB), TB, 0, stream>>>(h1, att_src1, att_dst1, asrc1, adst1,
                                                 (int)N, H1, C1);
  edge_max<<<cdiv(ET * H1, TB), TB, 0, stream>>>(ei, E, ET, H1, asrc1, adst1, m1);
  edge_sum<<<cdiv(ET * H1, TB), TB, 0, stream>>>(ei, E, ET, H1, asrc1, adst1, m1, den1);
  edge_aggregate<F1, C1><<<cdiv(ET, 256 / F1), TB, 0, stream>>>(ei, E, ET, H1, asrc1, adst1,
                                                                m1, den1, h1, out1);
  elu_bias<<<cdiv(N * F1, TB), TB, 0, stream>>>(out1, b1, N * F1, F1);

  // ---- layer 2 ----
  {
    long long waves = N / 16;
    size_t shmem = (size_t)(KCHUNK / 2) * F2 * sizeof(v2f); // 32KB
    gemm_wmma_f32_lds<4><<<cdiv(waves, TB / 32), TB, shmem, stream>>>(
        out1, W2, h2, (int)N, F1, F2, KCHUNK);
  }
  att_coeff<<<cdiv(N * H2, TB), TB, 0, stream>>>(h2, att_src2, att_dst2, asrc2, adst2,
                                                 (int)N, H2, C2);
  edge_max<<<cdiv(ET * H2, TB), TB, 0, stream>>>(ei, E, ET, H2, asrc2, adst2, m2);
  edge_sum<<<cdiv(ET * H2, TB), TB, 0, stream>>>(ei, E, ET, H2, asrc2, adst2, m2, den2);
  edge_aggregate<F2, C2><<<cdiv(ET, 256 / F2), TB, 0, stream>>>(ei, E, ET, H2, asrc2, adst2,
                                                                m2, den2, h2, out);
  // mean over H2=1 heads is identity; just add bias
  add_bias<<<cdiv(N * F2, TB), TB, 0, stream>>>(out, b2, N * F2, F2);
}